// MultipleChoiceLoss_49374944035068
// MI455X (gfx1250) — compile-verified
//
#include <hip/hip_runtime.h>
#include <hip/hip_bf16.h>

typedef __attribute__((ext_vector_type(2))) float v2f;
typedef __attribute__((ext_vector_type(8))) float v8f;

// Full 32-lane sum using V_WMMA_F32_16X16X4_F32.
// Stage 1: A[m][0]=x[m], A[m][2]=x[m+16] (documented 16x4 f32 A layout), B=ones
//          -> D[m][n] = x[m]+x[m+16] = p[m]  (independent of B layout since B==1)
//          D layout: vgpr j: lanes0-15 = p[j], lanes16-31 = p[j+8]
// Stage 2: feed D vgpr pairs back as A against B=ones, C-chained 4x:
//          total = sum_{j=0..15} p[j] = sum of all 32 x, in every lane.
__device__ __forceinline__ float wave_sum32_wmma(float x) {
    v2f a;    a[0] = x;    a[1] = 0.0f;
    v2f ones; ones[0] = 1.0f; ones[1] = 1.0f;
    v8f z = {};
    v8f d1 = __builtin_amdgcn_wmma_f32_16x16x4_f32(
        false, a, false, ones, (short)0, z, false, false);
    v8f acc = {};
#pragma unroll
    for (int j = 0; j < 4; ++j) {
        v2f a2; a2[0] = d1[2 * j]; a2[1] = d1[2 * j + 1];
        acc = __builtin_amdgcn_wmma_f32_16x16x4_f32(
            false, a2, false, ones, (short)0, acc, false, false);
    }
    return acc[0];
}

// One block per row: async-copy row -> LDS, hinge-sum, write partial sum.
__global__ void __launch_bounds__(256)
hinge_row_kernel(const float* __restrict__ score,
                 const int* __restrict__ target,
                 float* __restrict__ partials,
                 int C) {
    extern __shared__ float lds_row[];
    __shared__ float wsum[8];

    const int row = blockIdx.x;
    const int tid = threadIdx.x;
    const unsigned long long rbase =
        (unsigned long long)(score + (size_t)row * (size_t)C);

    const int C16 = C >> 2;  // number of 16-byte chunks

    // Async global -> LDS streaming copy (ASYNCcnt path, gfx1250).
    for (int i = tid; i < C16; i += 256) {
        unsigned lds_addr = (unsigned)(size_t)(&lds_row[i * 4]);
        unsigned goff     = (unsigned)(i * 16);
        asm volatile("global_load_async_to_lds_b128 %0, %1, %2"
                     :
                     : "v"(lds_addr), "v"(goff), "s"(rbase)
                     : "memory");
    }
    for (int c = (C16 << 2) + tid; c < C; c += 256) {  // scalar tail
        unsigned lds_addr = (unsigned)(size_t)(&lds_row[c]);
        unsigned goff     = (unsigned)(c * 4);
        asm volatile("global_load_async_to_lds_b32 %0, %1, %2"
                     :
                     : "v"(lds_addr), "v"(goff), "s"(rbase)
                     : "memory");
    }
    asm volatile("s_wait_asynccnt 0" ::: "memory");
    __syncthreads();

    // Gather the "correct" score from LDS (cheap, no divergent global gather).
    const int   t    = target[row];
    const float bias = 1.0f - lds_row[t];  // hinge = max(0, s + (1 - correct))

    float acc = 0.0f;
    const float4* r4 = reinterpret_cast<const float4*>(lds_row);
    for (int i = tid; i < C16; i += 256) {
        float4 v = r4[i];
        acc += fmaxf(0.0f, v.x + bias) + fmaxf(0.0f, v.y + bias) +
               fmaxf(0.0f, v.z + bias) + fmaxf(0.0f, v.w + bias);
    }
    for (int c = (C16 << 2) + tid; c < C; c += 256)
        acc += fmaxf(0.0f, lds_row[c] + bias);

    // Wave reduce via WMMA, then cross-wave via LDS + one more WMMA reduce.
    float wtotal = wave_sum32_wmma(acc);
    const int wid = tid >> 5, lane = tid & 31;
    if (lane == 0) wsum[wid] = wtotal;
    __syncthreads();
    if (wid == 0) {
        float v   = (lane < 8) ? wsum[lane] : 0.0f;
        float tot = wave_sum32_wmma(v);
        // c==target contributes exactly max(0, 1+0) = 1.0 -> subtract it.
        if (lane == 0) partials[row] = tot - 1.0f;
    }
}

// Single-block deterministic final reduction (fixed order, no atomics).
__global__ void __launch_bounds__(256)
finalize_kernel(const float* __restrict__ partials,
                float* __restrict__ out,
                int N, float inv_cnt) {
    __shared__ float wsum[8];
    const int tid = threadIdx.x;
    float acc = 0.0f;
    for (int i = tid; i < N; i += 256) acc += partials[i];
    float wtotal = wave_sum32_wmma(acc);
    const int wid = tid >> 5, lane = tid & 31;
    if (lane == 0) wsum[wid] = wtotal;
    __syncthreads();
    if (wid == 0) {
        float v   = (lane < 8) ? wsum[lane] : 0.0f;
        float tot = wave_sum32_wmma(v);
        if (lane == 0) out[0] = tot * inv_cnt;
    }
}

extern "C" void kernel_launch(void* const* d_in, const int* in_sizes, int n_in,
                              void* d_out, int out_size, void* d_ws, size_t ws_size,
                              hipStream_t stream) {
    const float* score  = (const float*)d_in[0];
    const int*   target = (const int*)d_in[1];
    const int N = in_sizes[1];
    const int C = in_sizes[0] / N;

    float* partials = (float*)d_ws;  // N floats of scratch

    size_t shmem = (size_t)((C * 4 + 15) & ~15);  // row bytes, 16B aligned
    hinge_row_kernel<<<N, 256, shmem, stream>>>(score, target, partials, C);

    const float inv_cnt = 1.0f / (float)((long long)N * (long long)(C - 1));
    finalize_kernel<<<1, 256, 0, stream>>>(partials, (float*)d_out, N, inv_cnt);
}